// DynamicGraphConstructor_5222680232053
// MI455X (gfx1250) — compile-verified
//
#include <hip/hip_runtime.h>
#include <hip/hip_bf16.h>

typedef __attribute__((ext_vector_type(2))) float v2f;
typedef __attribute__((ext_vector_type(8))) float v8f;

#define NNODES 4096
#define CDIM   64
#define KNN    8
#define BIGF   3.402823466e38f

// ---------------------------------------------------------------------------
// Kernel 1: projection  nodes[b,n,o] = sum_c feat[b,c,n] * W[o,c] + bias[o]
// also emits xx[b*N+n] = ||nodes[b,n,:]||^2 into workspace.
// Bandwidth-bound (~16MB traffic). W (16KB) staged in LDS, feature row in regs.
// ---------------------------------------------------------------------------
__global__ __launch_bounds__(256) void proj_kernel(
    const float* __restrict__ feat, const float* __restrict__ W,
    const float* __restrict__ bias, float* __restrict__ nodes,
    float* __restrict__ xx) {
  __shared__ float sW[CDIM * CDIM];
  const int tid = threadIdx.x;
  for (int i = tid; i < CDIM * CDIM; i += 256) sW[i] = W[i];
  __syncthreads();

  const int p = blockIdx.x * 256 + tid;         // 0 .. 32767
  const int b = p >> 12;
  const int n = p & (NNODES - 1);
  const float* fb = feat + (size_t)b * CDIM * NNODES + n;

  float f[CDIM];
#pragma unroll
  for (int c = 0; c < CDIM; ++c) f[c] = fb[c * NNODES];

  float* np = nodes + (size_t)p * CDIM;
  float xv = 0.0f;
  for (int o = 0; o < CDIM; ++o) {
    float acc = bias[o];
#pragma unroll
    for (int c = 0; c < CDIM; ++c) acc += f[c] * sW[o * CDIM + c];
    np[o] = acc;
    xv += acc * acc;
  }
  xx[p] = xv;
}

// ---------------------------------------------------------------------------
// Kernel 2: fused Gram (WMMA f32 16x16x4) + distance + streaming top-8.
// One wave32 per 16-row strip per batch. A strip cached in regs (WMMA A
// layout: lane%16 = M row, VGPR pair = K slice, lane/16 selects K+2).
// B tiles stream per 16-column block (L2 resident, per-batch X = 1MB).
// C layout: VGPR r, lanes 0-15 -> row r, lanes 16-31 -> row r+8; col = lane%16.
// Each lane keeps a sorted top-8 (dist,idx) list per owned row; merge via LDS.
// ---------------------------------------------------------------------------
__global__ __launch_bounds__(32) void knn_kernel(
    const float* __restrict__ nodes, const float* __restrict__ xx,
    float* __restrict__ edge_out) {
  const int lane  = threadIdx.x;       // 0..31
  const int half  = lane >> 4;         // 0/1
  const int mlane = lane & 15;
  const int b     = blockIdx.y;
  const int i0    = blockIdx.x * 16;

  const float* X   = nodes + (size_t)b * NNODES * CDIM;
  const float* xxb = xx + b * NNODES;

  // A strip: rows i0..i0+15, K=64 as 16 WMMA K-steps of 4.
  v2f a[16];
#pragma unroll
  for (int kk = 0; kk < 16; ++kk)
    a[kk] = *(const v2f*)(X + (size_t)(i0 + mlane) * CDIM + kk * 4 + 2 * half);

  float xxi[8];
#pragma unroll
  for (int r = 0; r < 8; ++r) xxi[r] = xxb[i0 + r + 8 * half];

  float bd[8][KNN];
  int   bi[8][KNN];
#pragma unroll
  for (int r = 0; r < 8; ++r)
#pragma unroll
    for (int s = 0; s < KNN; ++s) { bd[r][s] = BIGF; bi[r][s] = 0; }

  for (int j0 = 0; j0 < NNODES; j0 += 16) {
    v8f c = {};
#pragma unroll
    for (int kk = 0; kk < 16; ++kk) {
      v2f bv = *(const v2f*)(X + (size_t)(j0 + mlane) * CDIM + kk * 4 + 2 * half);
      // D = A(16x4) * B(4x16) + C  -- fp32 WMMA
      c = __builtin_amdgcn_wmma_f32_16x16x4_f32(
          /*neg_a=*/false, a[kk], /*neg_b=*/false, bv,
          /*c_mod=*/(short)0, c, /*reuse_a=*/false, /*reuse_b=*/false);
    }
    const float xxj = xxb[j0 + mlane];
    const int   col = j0 + mlane;
#pragma unroll
    for (int r = 0; r < 8; ++r) {
      const int row = i0 + r + 8 * half;
      float d = xxi[r] - 2.0f * c[r] + xxj;
      if (col == row) d = BIGF;            // drop self (ref drops nearest=self)
      if (d < bd[r][KNN - 1]) {            // branch-free sorted insert
        float dd = d; int ii = col;
#pragma unroll
        for (int s = 0; s < KNN; ++s) {
          const bool sw = dd < bd[r][s];
          const float td = sw ? bd[r][s] : dd;
          const int   ti = sw ? bi[r][s] : ii;
          bd[r][s] = sw ? dd : bd[r][s];
          bi[r][s] = sw ? ii : bi[r][s];
          dd = td; ii = ti;
        }
      }
    }
  }

  // Merge 16 per-lane partial lists per row through LDS, select top-8.
  __shared__ float s_d[16][16 * KNN];
  __shared__ int   s_i[16][16 * KNN];
#pragma unroll
  for (int r = 0; r < 8; ++r)
#pragma unroll
    for (int s = 0; s < KNN; ++s) {
      s_d[r + 8 * half][mlane * KNN + s] = bd[r][s];
      s_i[r + 8 * half][mlane * KNN + s] = bi[r][s];
    }
  __syncthreads();

  if (lane < 16) {
    const int row = i0 + lane;
    const size_t ebase = (size_t)b * (2 * NNODES * KNN);
    for (int t = 0; t < KNN; ++t) {
      float best = BIGF; int besti = 0, bestq = 0;
      for (int q = 0; q < 16 * KNN; ++q) {
        const float v = s_d[lane][q];
        if (v < best) { best = v; besti = s_i[lane][q]; bestq = q; }
      }
      s_d[lane][bestq] = BIGF;
      edge_out[ebase + (size_t)row * KNN + t]                = (float)row;
      edge_out[ebase + NNODES * KNN + (size_t)row * KNN + t] = (float)besti;
    }
  }
}

extern "C" void kernel_launch(void* const* d_in, const int* in_sizes, int n_in,
                              void* d_out, int out_size, void* d_ws, size_t ws_size,
                              hipStream_t stream) {
  const float* feat = (const float*)d_in[0];   // (8,64,64,64) f32
  const float* W    = (const float*)d_in[1];   // (64,64) f32
  const float* bias = (const float*)d_in[2];   // (64,) f32
  // d_in[3] = k (==8), hardcoded.

  float* out   = (float*)d_out;
  float* nodes = out;                          // 8*4096*64 floats
  float* edges = out + (size_t)8 * NNODES * CDIM;
  float* xx    = (float*)d_ws;                 // 8*4096 floats scratch

  proj_kernel<<<dim3(32768 / 256), 256, 0, stream>>>(feat, W, bias, nodes, xx);
  knn_kernel<<<dim3(NNODES / 16, 8), 32, 0, stream>>>(nodes, xx, edges);
}